// GATActorCritic_4750233829756
// MI455X (gfx1250) — compile-verified
//
#include <hip/hip_runtime.h>

// ---------------------------------------------------------------------------
// GAT actor-critic for MI455X (gfx1250).
//  - Dense GEMMs: V_WMMA_F32_16X16X4_F32 (full fp32 matrix pipe), B staged in
//    LDS per block (8 waves share it) via async global->LDS copy
//    (GLOBAL_LOAD_ASYNC_TO_LDS_B32, ASYNCcnt path), ds fallback otherwise.
//  - Attention segment-softmax / aggregation via L2-resident float atomics.
// ---------------------------------------------------------------------------

typedef __attribute__((ext_vector_type(2))) float v2f;
typedef __attribute__((ext_vector_type(8))) float v8f;

#define IN_DIM 128
#define HID    32
#define HEADS1 4
#define AGENTS 32
#define NEG_SLOPE 0.2f

typedef __attribute__((address_space(1))) int glob_i32;
typedef __attribute__((address_space(3))) int lds_i32;

// -------- async global->LDS staging support detection ----------------------
#ifdef __has_builtin
#  if __has_builtin(__builtin_amdgcn_global_load_async_to_lds_b32)
#    define ASYNC_STAGE 1
#  endif
#endif
#ifndef ASYNC_STAGE
#  define ASYNC_STAGE 0
#endif

#if ASYNC_STAGE
static __device__ __forceinline__ void wait_async0() {
#  if __has_builtin(__builtin_amdgcn_s_wait_asynccnt)
    __builtin_amdgcn_s_wait_asynccnt(0);
#  else
    asm volatile("s_wait_asynccnt 0x0" ::: "memory");
#  endif
}
#endif

static __device__ __forceinline__ unsigned f32_ord(float f) {
    unsigned u = __float_as_uint(f);
    return (u & 0x80000000u) ? ~u : (u | 0x80000000u);
}
static __device__ __forceinline__ float ord_f32(unsigned k) {
    return (k & 0x80000000u) ? __uint_as_float(k ^ 0x80000000u)
                             : __uint_as_float(~k);
}

// ------------------------- generic zero fill -------------------------------
__global__ void fill_zero_u32(unsigned* p, long long n) {
    long long i = (long long)blockIdx.x * blockDim.x + threadIdx.x;
    if (i < n) p[i] = 0u;
}

// ------------------------- fp32 WMMA GEMM, B in LDS ------------------------
// C[M,Nn] = A[M,K] @ B[K,Nn], row-major, ldb == Nn.
// Requires M%16==0, Nn%16==0, K%4==0. Dynamic LDS = K*Nn*4 bytes (<= 64 KB).
// One wave computes one 16x16 tile with V_WMMA_F32_16X16X4_F32; 8 waves per
// block share the LDS-resident B.
__global__ void wmma_gemm_f32_lds(const float* __restrict__ A,
                                  const float* __restrict__ B,
                                  float* __restrict__ C,
                                  int M, int Nn, int K, int lda, int ldc) {
    extern __shared__ float Bs[];                 // [K, Nn]
    const int tid   = threadIdx.x;
    const int total = K * Nn;

#if ASYNC_STAGE
    for (int i = tid; i < total; i += blockDim.x)
        __builtin_amdgcn_global_load_async_to_lds_b32(
            (glob_i32*)(B + i), (lds_i32*)(Bs + i), 0, 0);
    wait_async0();
    __syncthreads();
#else
    for (int i = tid; i < total; i += blockDim.x) Bs[i] = B[i];
    __syncthreads();
#endif

    const int nWaves = blockDim.x >> 5;
    const int wave   = blockIdx.x * nWaves + (tid >> 5);
    const int lane   = tid & 31;
    const int tilesN = Nn >> 4;
    const int tiles  = (M >> 4) * tilesN;
    if (wave >= tiles) return;            // wave-uniform: EXEC stays all-ones

    const int tm = wave / tilesN;
    const int tn = wave - tm * tilesN;
    const int mrow  = tm * 16 + (lane & 15);
    const int ncol  = tn * 16 + (lane & 15);
    const int khalf = (lane >> 4) * 2;    // lanes 16..31 handle K+2,K+3

    v8f acc = {};
    for (int k0 = 0; k0 < K; k0 += 4) {
        const float* ap = A + (size_t)mrow * lda + (k0 + khalf);
        v2f a = { ap[0], ap[1] };
        const float* bp = Bs + (k0 + khalf) * Nn + ncol;
        v2f b = { bp[0], bp[Nn] };
        acc = __builtin_amdgcn_wmma_f32_16x16x4_f32(
            /*neg_a=*/false, a, /*neg_b=*/false, b,
            /*c_mod=*/(short)0, acc, /*reuse_a=*/false, /*reuse_b=*/false);
    }
    // D layout: VGPR r, lanes 0-15 -> M=r, lanes 16-31 -> M=r+8, N=lane&15
    const int mbase = tm * 16 + ((lane >> 4) << 3);
#pragma unroll
    for (int r = 0; r < 8; ++r)
        C[(size_t)(mbase + r) * ldc + ncol] = acc[r];
}

// ------------------ per-node attention score dots --------------------------
__global__ void node_scores(const float* __restrict__ H,
                            const float* __restrict__ a_src,
                            const float* __restrict__ a_dst,
                            float* __restrict__ as, float* __restrict__ ad,
                            int N, int heads, int F) {
    int idx = blockIdx.x * blockDim.x + threadIdx.x;   // n*heads + h
    if (idx >= N * heads) return;
    int n = idx / heads, h = idx - n * heads;
    const float* hp = H + (size_t)n * heads * F + (size_t)h * F;
    float s = 0.f, d = 0.f;
    for (int f = 0; f < F; ++f) {
        s += hp[f] * a_src[h * F + f];
        d += hp[f] * a_dst[h * F + f];
    }
    as[idx] = s;
    ad[idx] = d;
}

// -------- pass 1: edge logits (leaky relu) + segment max (atomic) ----------
__global__ void edge_logits(const int* __restrict__ ei, int E, int N, int heads,
                            const float* __restrict__ as,
                            const float* __restrict__ ad,
                            float* __restrict__ ebuf,
                            unsigned* __restrict__ mkey) {
    int etot = E + N;
    int idx = blockIdx.x * blockDim.x + threadIdx.x;   // e*heads + h
    if (idx >= etot * heads) return;
    int e = idx / heads, h = idx - e * heads;
    int s, d;
    if (e < E) { s = ei[e]; d = ei[E + e]; } else { s = d = e - E; }
    float v = as[s * heads + h] + ad[d * heads + h];
    v = v > 0.f ? v : NEG_SLOPE * v;
    ebuf[idx] = v;
    atomicMax(&mkey[d * heads + h], f32_ord(v));
}

// -------- pass 2: exp(e - m), segment-sum denominator ----------------------
__global__ void edge_exp(const int* __restrict__ ei, int E, int N, int heads,
                         float* __restrict__ ebuf,
                         const unsigned* __restrict__ mkey,
                         float* __restrict__ den) {
    int etot = E + N;
    int idx = blockIdx.x * blockDim.x + threadIdx.x;
    if (idx >= etot * heads) return;
    int e = idx / heads, h = idx - e * heads;
    int d = (e < E) ? ei[E + e] : (e - E);
    float m = ord_f32(mkey[d * heads + h]);
    float ex = __expf(ebuf[idx] - m);
    ebuf[idx] = ex;
    atomicAdd(&den[d * heads + h], ex);
}

// -------- pass 3: alpha-weighted scatter-add of source features ------------
__global__ void edge_aggregate(const int* __restrict__ ei, int E, int N,
                               int heads, int F,
                               const float* __restrict__ ebuf,
                               const float* __restrict__ den,
                               const float* __restrict__ H,
                               float* __restrict__ agg) {
    int etot = E + N;
    int idx = blockIdx.x * blockDim.x + threadIdx.x;   // e*heads + h
    if (idx >= etot * heads) return;
    int e = idx / heads, h = idx - e * heads;
    int s, d;
    if (e < E) { s = ei[e]; d = ei[E + e]; } else { s = d = e - E; }
    float alpha = ebuf[idx] / den[d * heads + h];
    const float* hp = H + (size_t)s * heads * F + (size_t)h * F;
    float* op = agg + (size_t)d * heads * F + (size_t)h * F;
    for (int f = 0; f < F; ++f)
        atomicAdd(&op[f], alpha * hp[f]);
}

// ------------------ bias + ELU (in place) ----------------------------------
__global__ void bias_elu(float* __restrict__ y, const float* __restrict__ b,
                         long long total, int C) {
    long long i = (long long)blockIdx.x * blockDim.x + threadIdx.x;
    if (i >= total) return;
    float v = y[i] + b[(int)(i % C)];
    y[i] = v > 0.f ? v : (__expf(v) - 1.0f);
}

// ------------------ actor head: transpose + bias ---------------------------
__global__ void actor_store(const float* __restrict__ act,
                            const float* __restrict__ ba,
                            float* __restrict__ out, int N) {
    int i = blockIdx.x * blockDim.x + threadIdx.x;     // n*AGENTS + a
    if (i >= N * AGENTS) return;
    int n = i / AGENTS, a = i - n * AGENTS;
    out[(size_t)a * N + n] = act[(size_t)n * AGENTS + a] + ba[a];
}

// ------------------ critic head: block-reduced mean over nodes -------------
__global__ void critic_mean(const float* __restrict__ h,
                            const float* __restrict__ Wc,
                            const float* __restrict__ bc,
                            float* __restrict__ val, int N) {
    __shared__ float red[256];
    int n = blockIdx.x * blockDim.x + threadIdx.x;
    float acc = 0.f;
    if (n < N) {
        acc = bc[0];
        for (int f = 0; f < HID; ++f) acc += h[(size_t)n * HID + f] * Wc[f];
    }
    red[threadIdx.x] = acc;
    __syncthreads();
    for (int s = blockDim.x >> 1; s > 0; s >>= 1) {
        if (threadIdx.x < (unsigned)s) red[threadIdx.x] += red[threadIdx.x + s];
        __syncthreads();
    }
    if (threadIdx.x == 0) atomicAdd(val, red[0] / (float)N);
}

// ---------------------------------------------------------------------------
static inline unsigned cdiv(long long n, int b) {
    return (unsigned)((n + b - 1) / b);
}

extern "C" void kernel_launch(void* const* d_in, const int* in_sizes, int n_in,
                              void* d_out, int out_size, void* d_ws, size_t ws_size,
                              hipStream_t stream) {
    const float* x   = (const float*)d_in[0];
    const int*   ei  = (const int*)  d_in[1];   // edge_index [2,E]
    const float* W1  = (const float*)d_in[2];
    const float* a1s = (const float*)d_in[3];
    const float* a1d = (const float*)d_in[4];
    const float* b1  = (const float*)d_in[5];
    const float* W2  = (const float*)d_in[6];
    const float* a2s = (const float*)d_in[7];
    const float* a2d = (const float*)d_in[8];
    const float* b2  = (const float*)d_in[9];
    const float* Wa  = (const float*)d_in[10];
    const float* ba  = (const float*)d_in[11];
    const float* Wc  = (const float*)d_in[12];
    const float* bc  = (const float*)d_in[13];
    float* out = (float*)d_out;

    const int N    = in_sizes[0] / IN_DIM;   // 50000 (multiple of 16)
    const int E    = in_sizes[1] / 2;        // 800000
    const int Etot = E + N;

    // ---------------- workspace layout (floats) ----------------------------
    float*    H1   = (float*)d_ws;                       // [N,128]; reused L2
    float*    AGG1 = H1   + (size_t)N * 128;             // [N,128]; reused ACT
    float*    EDGE = AGG1 + (size_t)N * 128;             // [Etot,4] / [Etot,1]
    float*    AS   = EDGE + (size_t)Etot * 4;            // [N,4]
    float*    AD   = AS   + (size_t)N * 4;               // [N,4]
    unsigned* MK   = (unsigned*)(AD + (size_t)N * 4);    // [N,4]
    float*    DEN  = (float*)(MK + (size_t)N * 4);       // [N,4]
    // layer-2 aliases into dead regions:
    float* H2   = H1;                     // [N,32]
    float* AGG2 = H1 + (size_t)N * HID;   // [N,32]
    float* ACT  = AGG1;                   // [N,32] actor GEMM output
    float* VAL  = out + (size_t)AGENTS * N;

    const int BLK = 256;
    const int GEMM_WAVES = 8;             // 256-thread blocks, 8 tiles each

    // =================== layer 1: GATConv(128 -> 4x32) =====================
    fill_zero_u32<<<cdiv((long long)N * 8, BLK), BLK, 0, stream>>>(MK, (long long)N * 8);
    fill_zero_u32<<<cdiv((long long)N * 128, BLK), BLK, 0, stream>>>((unsigned*)AGG1, (long long)N * 128);
    fill_zero_u32<<<1, 32, 0, stream>>>((unsigned*)VAL, 1);

    {   // H1 = x @ W1   [N,128]x[128,128], B in LDS (64 KB)
        long long tiles = (long long)(N / 16) * ((HEADS1 * HID) / 16);
        size_t lds = (size_t)IN_DIM * (HEADS1 * HID) * sizeof(float);
        wmma_gemm_f32_lds<<<cdiv(tiles, GEMM_WAVES), 32 * GEMM_WAVES, lds, stream>>>(
            x, W1, H1, N, HEADS1 * HID, IN_DIM, IN_DIM, HEADS1 * HID);
    }
    node_scores<<<cdiv((long long)N * HEADS1, BLK), BLK, 0, stream>>>(
        H1, a1s, a1d, AS, AD, N, HEADS1, HID);
    edge_logits<<<cdiv((long long)Etot * HEADS1, BLK), BLK, 0, stream>>>(
        ei, E, N, HEADS1, AS, AD, EDGE, MK);
    edge_exp<<<cdiv((long long)Etot * HEADS1, BLK), BLK, 0, stream>>>(
        ei, E, N, HEADS1, EDGE, MK, DEN);
    edge_aggregate<<<cdiv((long long)Etot * HEADS1, BLK), BLK, 0, stream>>>(
        ei, E, N, HEADS1, HID, EDGE, DEN, H1, AGG1);
    bias_elu<<<cdiv((long long)N * 128, BLK), BLK, 0, stream>>>(
        AGG1, b1, (long long)N * 128, HEADS1 * HID);

    // =================== layer 2: GATConv(128 -> 1x32) =====================
    {   // H2 = AGG1 @ W2   [N,128]x[128,32], B in LDS (16 KB)
        long long tiles = (long long)(N / 16) * (HID / 16);
        size_t lds = (size_t)(HEADS1 * HID) * HID * sizeof(float);
        wmma_gemm_f32_lds<<<cdiv(tiles, GEMM_WAVES), 32 * GEMM_WAVES, lds, stream>>>(
            AGG1, W2, H2, N, HID, HEADS1 * HID, HEADS1 * HID, HID);
    }
    fill_zero_u32<<<cdiv((long long)N, BLK), BLK, 0, stream>>>(MK, (long long)N);
    fill_zero_u32<<<cdiv((long long)N, BLK), BLK, 0, stream>>>((unsigned*)DEN, (long long)N);
    fill_zero_u32<<<cdiv((long long)N * HID, BLK), BLK, 0, stream>>>((unsigned*)AGG2, (long long)N * HID);

    node_scores<<<cdiv((long long)N, BLK), BLK, 0, stream>>>(
        H2, a2s, a2d, AS, AD, N, 1, HID);
    edge_logits<<<cdiv((long long)Etot, BLK), BLK, 0, stream>>>(
        ei, E, N, 1, AS, AD, EDGE, MK);
    edge_exp<<<cdiv((long long)Etot, BLK), BLK, 0, stream>>>(
        ei, E, N, 1, EDGE, MK, DEN);
    edge_aggregate<<<cdiv((long long)Etot, BLK), BLK, 0, stream>>>(
        ei, E, N, 1, HID, EDGE, DEN, H2, AGG2);
    bias_elu<<<cdiv((long long)N * HID, BLK), BLK, 0, stream>>>(
        AGG2, b2, (long long)N * HID, HID);

    // =================== actor / critic heads ==============================
    {   // ACT = h @ Wa   [N,32]x[32,32], B in LDS (4 KB)
        long long tiles = (long long)(N / 16) * (AGENTS / 16);
        size_t lds = (size_t)HID * AGENTS * sizeof(float);
        wmma_gemm_f32_lds<<<cdiv(tiles, GEMM_WAVES), 32 * GEMM_WAVES, lds, stream>>>(
            AGG2, Wa, ACT, N, AGENTS, HID, HID, AGENTS);
    }
    actor_store<<<cdiv((long long)N * AGENTS, BLK), BLK, 0, stream>>>(ACT, ba, out, N);
    critic_mean<<<cdiv((long long)N, BLK), BLK, 0, stream>>>(AGG2, Wc, bc, VAL, N);
}